// MoELayer_7894149890292
// MI455X (gfx1250) — compile-verified
//
#include <hip/hip_runtime.h>
#include <math.h>

#define N_TOK 4096
#define DIM   2048
#define NEXP  8
#define EDIM  2048

typedef float v2f __attribute__((ext_vector_type(2)));
typedef float v8f __attribute__((ext_vector_type(8)));

// ---------------------------------------------------------------- zero output
__global__ __launch_bounds__(256) void zero_out_kernel(float4* __restrict__ out) {
    size_t i = (size_t)blockIdx.x * blockDim.x + threadIdx.x;
    out[i] = make_float4(0.f, 0.f, 0.f, 0.f);
}

// ---------------------------------------------------------------- router
// One block per token: RMSNorm -> D x E matvec -> softmax -> top2 -> renorm.
__global__ __launch_bounds__(256) void router_kernel(
    const float* __restrict__ x, const float* __restrict__ rscale,
    const float* __restrict__ gate_w, int* __restrict__ idx_out,
    float* __restrict__ w_out)
{
    __shared__ float sred[NEXP * 256];
    const int n = blockIdx.x;
    const int tid = threadIdx.x;
    const float* xr = x + (size_t)n * DIM;

    // pass 1: sum of squares
    float ss = 0.f;
    for (int d = tid; d < DIM; d += 256) { float v = xr[d]; ss += v * v; }
    sred[tid] = ss; __syncthreads();
    for (int s = 128; s > 0; s >>= 1) {
        if (tid < s) sred[tid] += sred[tid + s];
        __syncthreads();
    }
    const float rms = rsqrtf(sred[0] / (float)DIM + 1e-6f) * rsqrtf((float)DIM);
    __syncthreads();

    // pass 2: logits
    float acc[NEXP];
    #pragma unroll
    for (int e = 0; e < NEXP; e++) acc[e] = 0.f;
    for (int d = tid; d < DIM; d += 256) {
        float hv = xr[d] * rms * rscale[d];
        const float* gw = gate_w + (size_t)d * NEXP;
        #pragma unroll
        for (int e = 0; e < NEXP; e++) acc[e] += hv * gw[e];
    }
    #pragma unroll
    for (int e = 0; e < NEXP; e++) sred[e * 256 + tid] = acc[e];
    __syncthreads();
    for (int s = 128; s > 0; s >>= 1) {
        if (tid < s) {
            #pragma unroll
            for (int e = 0; e < NEXP; e++) sred[e * 256 + tid] += sred[e * 256 + tid + s];
        }
        __syncthreads();
    }
    if (tid == 0) {
        float lg[NEXP], p[NEXP];
        float m = -1e30f;
        for (int e = 0; e < NEXP; e++) { lg[e] = sred[e * 256]; m = fmaxf(m, lg[e]); }
        float den = 0.f;
        for (int e = 0; e < NEXP; e++) { p[e] = expf(lg[e] - m); den += p[e]; }
        for (int e = 0; e < NEXP; e++) p[e] /= den;
        int i0 = 0;
        for (int e = 1; e < NEXP; e++) if (p[e] > p[i0]) i0 = e;      // lowest index on ties
        int i1 = (i0 == 0) ? 1 : 0;
        for (int e = 0; e < NEXP; e++) { if (e != i0 && p[e] > p[i1]) i1 = e; }
        float s2 = fmaxf(p[i0] + p[i1], 1e-12f);
        idx_out[n * 2 + 0] = i0; idx_out[n * 2 + 1] = i1;
        w_out[n * 2 + 0] = p[i0] / s2; w_out[n * 2 + 1] = p[i1] / s2;
    }
}

// ---------------------------------------------------------------- per-expert token lists
// Deterministic serial gather, one lane per expert; lists padded to mult of 16.
__global__ void build_lists_kernel(const int* __restrict__ idx,
                                   const float* __restrict__ wgt,
                                   const float* __restrict__ pes,
                                   int* __restrict__ counts,
                                   int* __restrict__ toks,
                                   float* __restrict__ wl)
{
    int e = threadIdx.x;
    if (e >= NEXP) return;
    float sc = pes[e];
    int cnt = 0;
    for (int n = 0; n < N_TOK; n++) {
        #pragma unroll
        for (int k = 0; k < 2; k++) {
            if (idx[n * 2 + k] == e) {
                toks[e * N_TOK + cnt] = n;
                wl[e * N_TOK + cnt] = wgt[n * 2 + k] * sc;
                cnt++;
            }
        }
    }
    while (cnt & 15) { toks[e * N_TOK + cnt] = 0; wl[e * N_TOK + cnt] = 0.f; cnt++; }
    counts[e] = cnt;
}

// ---------------------------------------------------------------- expert FFN (fp32 WMMA)
// 512 threads = 16 waves. One block handles 16 tokens of one expert.
// Wave w owns 128 gate cols + 128 up cols (phase A) and 128 D output cols (phase B).
// X tile is staged through a double-buffered LDS chunk so the global-load latency
// of chunk kc+1 overlaps the ~2K cycles of WMMA work on chunk kc.
#define NW     16
#define WCOLS  128
#define KC     128
#define LSTR   132   // LDS row stride (floats), avoids bank conflicts

__global__ __launch_bounds__(512) void expert_kernel(
    const float* __restrict__ x, const float* __restrict__ gate_up,
    const float* __restrict__ down, const int* __restrict__ counts,
    const int* __restrict__ toks, const float* __restrict__ wl,
    float* __restrict__ out)
{
    const int e    = blockIdx.y;
    const int tile = blockIdx.x;
    if (tile * 16 >= counts[e]) return;

    __shared__ float lds_x[2][16 * LSTR];
    __shared__ float lds_act[16 * LSTR];
    __shared__ int   lds_tok[16];
    __shared__ float lds_w[16];

    const int tid   = threadIdx.x;
    const int wave  = tid >> 5;        // 0..15 (also = staged X row)
    const int lane  = tid & 31;
    const int lrow  = lane & 15;       // row (A) or column (B/C) within 16
    const int khalf = (lane < 16) ? 0 : 2;  // f32 A/B frag: low lanes K{0,1}, high lanes K{2,3}

    if (tid < 16) {
        lds_tok[tid] = toks[e * N_TOK + tile * 16 + tid];
        lds_w[tid]   = wl[e * N_TOK + tile * 16 + tid];
    }
    __syncthreads();

    const float* gu_e = gate_up + (size_t)e * DIM * (2 * EDIM);
    const float* dn_e = down    + (size_t)e * EDIM * DIM;
    const float* xrow = x + (size_t)lds_tok[wave] * DIM + lane * 4;  // this thread's slice

    // ---------------- Phase A: hh = X_tile @ gate_up_e, gate+up columns per wave
    v8f accg[8], accu[8];
    #pragma unroll
    for (int t = 0; t < 8; t++) { accg[t] = (v8f)0.0f; accu[t] = (v8f)0.0f; }

    // prologue: stage chunk 0 into buffer 0
    {
        float4 v = *(const float4*)xrow;
        float* dp = &lds_x[0][wave * LSTR + lane * 4];
        dp[0] = v.x; dp[1] = v.y; dp[2] = v.z; dp[3] = v.w;
    }

    for (int kc = 0; kc < DIM / KC; kc++) {
        __syncthreads();   // buffer (kc&1) fully staged; prev readers of (kc&1)^1 done
        const int cur = kc & 1;

        // fetch next chunk into registers; latency overlaps this chunk's WMMAs
        float4 nxt;
        const bool have_next = (kc + 1) < (DIM / KC);
        if (have_next) nxt = *(const float4*)(xrow + (kc + 1) * KC);

        const float* bptr = gu_e + (size_t)(kc * KC + khalf) * (2 * EDIM) + wave * WCOLS + lrow;
        const float* xbase = &lds_x[cur][lrow * LSTR + khalf];
        for (int kk = 0; kk < KC; kk += 4) {
            v2f a;
            a.x = xbase[kk]; a.y = xbase[kk + 1];
            #pragma unroll
            for (int t = 0; t < 8; t++) {
                v2f bg, bu;
                bg.x = bptr[t * 16];
                bg.y = bptr[t * 16 + 2 * EDIM];           // K row +1
                bu.x = bptr[t * 16 + EDIM];               // up half
                bu.y = bptr[t * 16 + EDIM + 2 * EDIM];
                accg[t] = __builtin_amdgcn_wmma_f32_16x16x4_f32(
                    false, a, false, bg, (short)0, accg[t], false, false);
                accu[t] = __builtin_amdgcn_wmma_f32_16x16x4_f32(
                    false, a, false, bu, (short)0, accu[t], false, false);
            }
            bptr += (size_t)4 * (2 * EDIM);
        }

        if (have_next) {   // stage next chunk into the other buffer (disjoint from readers)
            float* dp = &lds_x[cur ^ 1][wave * LSTR + lane * 4];
            dp[0] = nxt.x; dp[1] = nxt.y; dp[2] = nxt.z; dp[3] = nxt.w;
        }
    }

    // ---------------- gating: act = gelu_exact(gate) * up  (in registers)
    #pragma unroll
    for (int t = 0; t < 8; t++) {
        #pragma unroll
        for (int v = 0; v < 8; v++) {
            float g = accg[t][v];
            float u = accu[t][v];
            float a = 0.5f * g * (1.0f + erff(g * 0.70710678118654752f));
            accg[t][v] = a * u;
        }
    }

    // ---------------- Phase B: out_cols(wave) += act @ down_e, chunked over ED
    v8f acco[8];
    #pragma unroll
    for (int t = 0; t < 8; t++) acco[t] = (v8f)0.0f;

    for (int kc = 0; kc < EDIM / KC; kc++) {
        if (wave == kc) {
            // stage this wave's act columns (C/D layout: vgpr v, lane half -> M)
            #pragma unroll
            for (int t = 0; t < 8; t++) {
                #pragma unroll
                for (int v = 0; v < 8; v++) {
                    int M = v + ((lane < 16) ? 0 : 8);
                    lds_act[M * LSTR + t * 16 + lrow] = accg[t][v];
                }
            }
        }
        __syncthreads();

        const float* bptr = dn_e + (size_t)(kc * KC + khalf) * DIM + wave * WCOLS + lrow;
        const float* abase = &lds_act[lrow * LSTR + khalf];
        for (int kk = 0; kk < KC; kk += 4) {
            v2f a;
            a.x = abase[kk]; a.y = abase[kk + 1];
            #pragma unroll
            for (int t = 0; t < 8; t++) {
                v2f b;
                b.x = bptr[t * 16];
                b.y = bptr[t * 16 + DIM];
                acco[t] = __builtin_amdgcn_wmma_f32_16x16x4_f32(
                    false, a, false, b, (short)0, acco[t], false, false);
            }
            bptr += (size_t)4 * DIM;
        }
        __syncthreads();
    }

    // ---------------- epilogue: scale by routing weight, atomic add (2 contribs/elem)
    #pragma unroll
    for (int t = 0; t < 8; t++) {
        #pragma unroll
        for (int v = 0; v < 8; v++) {
            int M = v + ((lane < 16) ? 0 : 8);
            float val = acco[t][v] * lds_w[M];
            int tok = lds_tok[M];
            unsafeAtomicAdd(&out[(size_t)tok * DIM + wave * WCOLS + t * 16 + lrow], val);
        }
    }
}

// ---------------------------------------------------------------- launch
extern "C" void kernel_launch(void* const* d_in, const int* in_sizes, int n_in,
                              void* d_out, int out_size, void* d_ws, size_t ws_size,
                              hipStream_t stream)
{
    const float* x       = (const float*)d_in[0];
    const float* rscale  = (const float*)d_in[1];
    const float* gate_w  = (const float*)d_in[2];
    const float* gate_up = (const float*)d_in[3];
    const float* down    = (const float*)d_in[4];
    const float* pes     = (const float*)d_in[5];
    float* out = (float*)d_out;

    char* ws = (char*)d_ws;
    int*   idx    = (int*)ws;                                    // N*2 ints
    float* wgt    = (float*)(ws + N_TOK * 2 * 4);                // N*2 floats
    int*   counts = (int*)(ws + N_TOK * 4 * 4);                  // E ints
    int*   toks   = (int*)(ws + N_TOK * 4 * 4 + 256);            // E*N ints
    float* wl     = (float*)(ws + N_TOK * 4 * 4 + 256 + NEXP * N_TOK * 4);

    zero_out_kernel<<<dim3((N_TOK * DIM) / (256 * 4)), 256, 0, stream>>>((float4*)out);
    router_kernel<<<dim3(N_TOK), 256, 0, stream>>>(x, rscale, gate_w, idx, wgt);
    build_lists_kernel<<<1, NEXP, 0, stream>>>(idx, wgt, pes, counts, toks, wl);
    expert_kernel<<<dim3(N_TOK / 16, NEXP), 512, 0, stream>>>(x, gate_up, down,
                                                              counts, toks, wl, out);
}